// EuclideanTriple_35132832481765
// MI455X (gfx1250) — compile-verified
//
#include <hip/hip_runtime.h>
#include <hip/hip_bf16.h>
#include <cstdint>

typedef __attribute__((ext_vector_type(2))) float v2f;
typedef __attribute__((ext_vector_type(8))) float v8f;

#define TMK_EPS    1e-6f
#define TMK_MARGIN 0.5f

// Geometry (reference fixes D = 256)
#define D_DIM      256
#define TILE_ROWS  16
#define K_STAGE    32                      // columns staged per buffer
#define N_STAGES   (D_DIM / K_STAGE)       // 8
#define ROW_F      (K_STAGE + 4)           // 36 floats/row: rows stay 16B-aligned for b128 LDS
                                           // writes AND ds_load_b64 reads are bank-conflict-free
#define TILE_F     (TILE_ROWS * ROW_F)     // 576 floats per array per buffer
#define TILE_B     (TILE_F * 4)            // 2304 bytes
#define BUF_F      (3 * TILE_F)            // 1728 floats (x, y, z tiles)
#define BUF_B      (BUF_F * 4)             // 6912 bytes
#define WAVE_F     (2 * BUF_F)             // 3456 floats: double buffered
#define WAVE_B     (WAVE_F * 4)            // 13824 bytes per wave
#define WAVES      4
#define BLOCK_T    (WAVES * 32)            // 128 threads = 4 wave32s

// One async global->LDS 16B-per-lane transfer (ASYNCcnt-tracked, CDNA5 path).
__device__ __forceinline__ void async_b128(const float* g, uint32_t lds_byte) {
  uint64_t ga = (uint64_t)(uintptr_t)g;
  asm volatile("global_load_async_to_lds_b128 %0, %1, off"
               :: "v"(lds_byte), "v"(ga)
               : "memory");
}

// Stage one 16x32 f32 tile of each of x,y,z into this wave's LDS buffer.
// 12 async b128 issues total (3 arrays x 4), fully coalesced: each issue moves
// four full 128B row-segments (32 lanes x 16B = 512B).
__device__ __forceinline__ void issue_stage(const float* x, const float* y, const float* z,
                                            long row0, int k0, uint32_t buf_base, int lane) {
#pragma unroll
  for (int i = 0; i < 4; ++i) {
    int chunk = i * 32 + lane;             // 0..127 : 16B chunk index in the 2KB tile
    int r     = chunk >> 3;                // row 0..15
    int c4    = (chunk & 7) << 2;          // float col within stage: 0,4,...,28
    size_t   goff = (size_t)(row0 + r) * D_DIM + (size_t)(k0 + c4);
    uint32_t loff = buf_base + (uint32_t)((r * ROW_F + c4) * 4);
    async_b128(x + goff, loff);
    async_b128(y + goff, loff + TILE_B);
    async_b128(z + goff, loff + 2 * TILE_B);
  }
}

__global__ __launch_bounds__(BLOCK_T)
void EuclideanTriple_35132832481765_kernel(const float* __restrict__ x,
                                           const float* __restrict__ y,
                                           const float* __restrict__ z,
                                           float* __restrict__ partials,
                                           int tiles) {
  __shared__ float smem[WAVES * WAVE_F];   // 55,296 B: per-wave disjoint regions
  __shared__ float wpart[WAVES];

  const int lane = threadIdx.x & 31;
  const int wave = threadIdx.x >> 5;
  const int m    = lane & 15;              // matrix row this lane feeds
  const int half = lane >> 4;              // K-phase pair selector (A 16x4 f32 layout)

  // Low 32 bits of the generic address of a __shared__ object == LDS byte offset.
  const uint32_t wave_base =
      (uint32_t)(uintptr_t)(void*)&smem[0] + (uint32_t)(wave * WAVE_B);

  const v2f ones = {1.0f, 1.0f};
  float wave_sum = 0.0f;

  for (int t = blockIdx.x * WAVES + wave; t < tiles; t += gridDim.x * WAVES) {
    const long row0 = (long)t * TILE_ROWS;

    // Prologue: stage 0 into buffer 0.
    issue_stage(x, y, z, row0, 0, wave_base, lane);

    // Per-lane K-phase partial sums of squared diffs (A-matrix fragments):
    // lane l<16 accumulates row l phases {0,1}; lane l>=16 row l-16 phases {2,3}.
    v2f accP = {0.0f, 0.0f};
    v2f accN = {0.0f, 0.0f};

    for (int s = 0; s < N_STAGES; ++s) {
      if (s + 1 < N_STAGES) {
        // Target buffer's previous ds_load reads must be drained before overwrite.
        asm volatile("s_wait_dscnt 0x0" ::: "memory");
        issue_stage(x, y, z, row0, (s + 1) * K_STAGE,
                    wave_base + (uint32_t)(((s + 1) & 1) * BUF_B), lane);
        // Allow only the 12 just-issued transfers to remain outstanding:
        // guarantees the current buffer's 12 have landed.
        asm volatile("s_wait_asynccnt 0xc" ::: "memory");
      } else {
        asm volatile("s_wait_asynccnt 0x0" ::: "memory");
      }

      const float*  wbuf = smem + wave * WAVE_F + (s & 1) * BUF_F;
      const float2* bx   = (const float2*)(wbuf);
      const float2* by   = (const float2*)(wbuf + TILE_F);
      const float2* bz   = (const float2*)(wbuf + 2 * TILE_F);

#pragma unroll
      for (int kk = 0; kk < K_STAGE / 4; ++kk) {
        const int f2 = m * (ROW_F / 2) + kk * 2 + half;   // conflict-free ds_load_b64
        float2 xv = bx[f2];
        float2 yv = by[f2];
        float2 zv = bz[f2];
        float dp0 = (xv.x - yv.x) + TMK_EPS;
        float dp1 = (xv.y - yv.y) + TMK_EPS;
        float dn0 = (xv.x - zv.x) + TMK_EPS;
        float dn1 = (xv.y - zv.y) + TMK_EPS;
        accP.x = fmaf(dp0, dp0, accP.x);
        accP.y = fmaf(dp1, dp1, accP.y);
        accN.x = fmaf(dn0, dn0, accN.x);
        accN.y = fmaf(dn1, dn1, accN.y);
      }
    }

    // Cross-K reduction via full-precision f32 WMMA: A = 16x4 phase partials,
    // B = ones  =>  D[m][n] = sum_j A[m][j] = ||row m||^2 in every column.
    v8f cz = {};
    v8f cP = __builtin_amdgcn_wmma_f32_16x16x4_f32(
        false, accP, false, ones, (short)0, cz, false, false);
    v8f cN = __builtin_amdgcn_wmma_f32_16x16x4_f32(
        false, accN, false, ones, (short)0, cz, false, false);

    // D layout: lanes 0-15 VGPR r = rows 0-7; lanes 16-31 VGPR r = rows 8-15.
    float local = 0.0f;
#pragma unroll
    for (int r = 0; r < 8; ++r) {
      float dP = sqrtf(cP[r]);
      float dN = sqrtf(cN[r]);
      local += fmaxf(dP + TMK_MARGIN - dN, 0.0f);
    }
    local += __shfl_xor(local, 16, 32);    // rows 0-7 sum + rows 8-15 sum
    wave_sum += local;                     // identical in every lane
  }

  if (lane == 0) wpart[wave] = wave_sum;
  __syncthreads();
  if (threadIdx.x == 0) {
    float s = 0.0f;
#pragma unroll
    for (int w = 0; w < WAVES; ++w) s += wpart[w];
    partials[blockIdx.x] = s;
  }
}

// Deterministic final reduction (fixed tree: bitwise-identical every replay).
__global__ __launch_bounds__(256)
void EuclideanTriple_reduce_kernel(const float* __restrict__ in, int n,
                                   float* __restrict__ out) {
  __shared__ float s[256];
  float a = 0.0f;
  for (int i = threadIdx.x; i < n; i += 256) a += in[i];
  s[threadIdx.x] = a;
  __syncthreads();
  for (int k = 128; k > 0; k >>= 1) {
    if ((int)threadIdx.x < k) s[threadIdx.x] += s[threadIdx.x + k];
    __syncthreads();
  }
  if (threadIdx.x == 0) out[0] = s[0];
}

extern "C" void kernel_launch(void* const* d_in, const int* in_sizes, int n_in,
                              void* d_out, int out_size, void* d_ws, size_t ws_size,
                              hipStream_t stream) {
  (void)n_in; (void)out_size;
  const float* x = (const float*)d_in[0];
  const float* y = (const float*)d_in[1];
  const float* z = (const float*)d_in[2];
  float* out = (float*)d_out;
  float* ws  = (float*)d_ws;

  const int N     = in_sizes[0] / D_DIM;       // 131072
  const int tiles = N / TILE_ROWS;             // 8192

  int blocks = (tiles + WAVES - 1) / WAVES;    // 2048 -> one tile per wave
  int max_blocks = (int)(ws_size / sizeof(float));
  if (max_blocks < 1) max_blocks = 1;
  if (blocks > max_blocks) blocks = max_blocks;
  if (blocks > 4096) blocks = 4096;
  if (blocks < 1) blocks = 1;

  EuclideanTriple_35132832481765_kernel<<<blocks, BLOCK_T, 0, stream>>>(x, y, z, ws, tiles);
  EuclideanTriple_reduce_kernel<<<1, 256, 0, stream>>>(ws, blocks, out);
}